// KWinners2d_30270929502270
// MI455X (gfx1250) — compile-verified
//
#include <hip/hip_runtime.h>
#include <stdint.h>
#include <math.h>

// Problem geometry (fixed by the reference setup)
#define B_   64
#define C_   128
#define HW_  3136                 // 56*56
#define N_   (C_ * HW_)           // 401408 elements per batch
#define N4_  (N_ / 4)             // 100352 float4 per batch
#define BINS 4096
#define TPB  256
#define SLICES 98                 // N4_ / (TPB * TILES)
#define V4_PER_BLOCK (N4_ / SLICES)   // 1024 float4 per block
#define TILES (V4_PER_BLOCK / TPB)    // 4 tiles of 256 float4

#ifndef __has_builtin
#define __has_builtin(x) 0
#endif

// gfx1250 async global->LDS path (guarded; falls back to direct loads)
#if defined(__HIP_DEVICE_COMPILE__) && __has_builtin(__builtin_amdgcn_global_load_async_to_lds_b128)
#define HAVE_ASYNC 1
#else
#define HAVE_ASYNC 0
#endif

#if HAVE_ASYNC
// Builtin prototype (from hipcc diagnostic): params are pointers to
// 'int __attribute__((vector_size(16)))' in addrspace(1) / addrspace(3).
typedef int v4i_ __attribute__((__vector_size__(16)));
typedef __attribute__((address_space(1))) v4i_* g_v4i_p;
typedef __attribute__((address_space(3))) v4i_* l_v4i_p;
#define ASYNC_CP16(gp, lp)                                                    \
    __builtin_amdgcn_global_load_async_to_lds_b128(                           \
        (g_v4i_p)(uintptr_t)(gp),                                             \
        (l_v4i_p)(unsigned)(uintptr_t)(lp),                                   \
        0, 0)
#if __has_builtin(__builtin_amdgcn_s_wait_asynccnt)
#define WAIT_ASYNC(n) __builtin_amdgcn_s_wait_asynccnt(n)
#else
#define WAIT_ASYNC(n) asm volatile("s_wait_asynccnt %0" ::"i"(n) : "memory")
#endif
#endif

// Order-preserving float -> uint mapping: a > b (float) <=> ordf(a) > ordf(b) (uint)
__device__ __forceinline__ unsigned ordf(float f) {
    unsigned u = __float_as_uint(f);
    return u ^ ((unsigned)(((int)u) >> 31) | 0x80000000u);
}

// ---------------------------------------------------------------------------
// Init: zero per-batch histograms, compute per-channel boost factors, seed state
__global__ void k_init(unsigned* __restrict__ hist, unsigned* __restrict__ prefix,
                       unsigned* __restrict__ needed, unsigned* __restrict__ tiecnt,
                       float* __restrict__ boost, const float* __restrict__ duty,
                       const int* __restrict__ kptr) {
    int gid = blockIdx.x * blockDim.x + threadIdx.x;
    if (gid < B_ * BINS) hist[gid] = 0u;
    if (gid < C_) {
        float td = (float)kptr[0] / (float)N_;
        boost[gid] = expf(td - duty[gid]);   // BOOST_STRENGTH == 1
    }
    if (gid < B_) {
        prefix[gid] = 0u;
        needed[gid] = (unsigned)kptr[0];
        tiecnt[gid] = 0u;
    }
}

// ---------------------------------------------------------------------------
// Level-0 histogram of ord(x*boost)[31:20]; streams x via async global->LDS
__global__ __launch_bounds__(TPB) void k_hist0(const float4* __restrict__ x4,
                                               const float* __restrict__ boost,
                                               unsigned* __restrict__ hist) {
    __shared__ unsigned h[BINS];
#if HAVE_ASYNC
    __shared__ float4 stage[2][TPB];
#endif
    const int tid = threadIdx.x;
    const int batch = blockIdx.y;
    const int slice = blockIdx.x;
    for (int i = tid; i < BINS; i += TPB) h[i] = 0u;
    __syncthreads();

    const float4* src = x4 + (size_t)batch * N4_ + (size_t)slice * V4_PER_BLOCK;

#if HAVE_ASYNC
    ASYNC_CP16(&src[tid], &stage[0][tid]);   // prologue: tile 0 in flight
#endif
    for (int t = 0; t < TILES; ++t) {
        float4 v;
#if HAVE_ASYNC
        if (t + 1 < TILES) {
            ASYNC_CP16(&src[(t + 1) * TPB + tid], &stage[(t + 1) & 1][tid]);
            WAIT_ASYNC(1);                   // oldest (current tile) complete
        } else {
            WAIT_ASYNC(0);
        }
        asm volatile("" ::: "memory");
        v = stage[t & 1][tid];               // each lane consumes its own slot
#else
        v = src[t * TPB + tid];
#endif
        int idx4 = slice * V4_PER_BLOCK + t * TPB + tid;   // float4 index in batch
        float bf = boost[idx4 / (HW_ / 4)];                // channel = elem / 3136
        atomicAdd(&h[ordf(v.x * bf) >> 20], 1u);
        atomicAdd(&h[ordf(v.y * bf) >> 20], 1u);
        atomicAdd(&h[ordf(v.z * bf) >> 20], 1u);
        atomicAdd(&h[ordf(v.w * bf) >> 20], 1u);
    }
    __syncthreads();
    unsigned* gh = hist + (size_t)batch * BINS;
    for (int i = tid; i < BINS; i += TPB) {
        unsigned c = h[i];
        if (c) atomicAdd(&gh[i], c);
    }
}

// ---------------------------------------------------------------------------
// Refinement histogram: only elements matching current prefix contribute
__global__ __launch_bounds__(TPB) void k_histL(const float4* __restrict__ x4,
                                               const float* __restrict__ boost,
                                               const unsigned* __restrict__ prefix,
                                               unsigned* __restrict__ hist,
                                               unsigned mask, int shift, unsigned binmask) {
    __shared__ unsigned h[BINS];
    const int tid = threadIdx.x;
    const int batch = blockIdx.y;
    const int slice = blockIdx.x;
    for (int i = tid; i < BINS; i += TPB) h[i] = 0u;
    __syncthreads();

    const unsigned pfx = prefix[batch];
    const float4* src = x4 + (size_t)batch * N4_ + (size_t)slice * V4_PER_BLOCK;
    for (int t = 0; t < TILES; ++t) {
        if (t + 1 < TILES) __builtin_prefetch(&src[(t + 1) * TPB + tid], 0, 1);
        float4 v = src[t * TPB + tid];
        int idx4 = slice * V4_PER_BLOCK + t * TPB + tid;
        float bf = boost[idx4 / (HW_ / 4)];
        unsigned o;
        o = ordf(v.x * bf); if ((o & mask) == pfx) atomicAdd(&h[(o >> shift) & binmask], 1u);
        o = ordf(v.y * bf); if ((o & mask) == pfx) atomicAdd(&h[(o >> shift) & binmask], 1u);
        o = ordf(v.z * bf); if ((o & mask) == pfx) atomicAdd(&h[(o >> shift) & binmask], 1u);
        o = ordf(v.w * bf); if ((o & mask) == pfx) atomicAdd(&h[(o >> shift) & binmask], 1u);
    }
    __syncthreads();
    unsigned* gh = hist + (size_t)batch * BINS;
    for (int i = tid; i < BINS; i += TPB) {
        unsigned c = h[i];
        if (c) atomicAdd(&gh[i], c);
    }
}

// ---------------------------------------------------------------------------
// Per-batch bin selection (descending scan); also clears histogram for reuse
__global__ void k_select(unsigned* __restrict__ hist, unsigned* __restrict__ prefix,
                         unsigned* __restrict__ needed, unsigned* __restrict__ thresh,
                         unsigned* __restrict__ tie_take, unsigned* __restrict__ tiecnt,
                         int nbins, int binshift, int final_level) {
    const int batch = blockIdx.x;
    unsigned* hb = hist + (size_t)batch * BINS;
    if (threadIdx.x == 0) {
        unsigned need = needed[batch];
        unsigned cum = 0;
        int b = nbins - 1;
        for (; b > 0; --b) {
            unsigned c = hb[b];
            if (cum + c >= need) break;
            cum += c;
        }
        if (final_level) {
            thresh[batch]   = prefix[batch] | (unsigned)b;  // exact 32-bit threshold
            tie_take[batch] = need - cum;                    // ties to keep at == T
            tiecnt[batch]   = 0u;
        } else {
            prefix[batch] |= ((unsigned)b) << binshift;
            needed[batch]  = need - cum;
        }
    }
    __syncthreads();
    for (int i = threadIdx.x; i < BINS; i += blockDim.x) hb[i] = 0u;
}

// ---------------------------------------------------------------------------
// Final write: keep original x where ord(x*boost) beats threshold
__device__ __forceinline__ float sel1(float v, float bf, unsigned T,
                                      unsigned take, unsigned* cnt) {
    unsigned o = ordf(v * bf);
    if (o > T) return v;
    if (o == T) {
        unsigned p = atomicAdd(cnt, 1u);     // rare: exact-threshold ties only
        return (p < take) ? v : 0.0f;
    }
    return 0.0f;
}

__global__ __launch_bounds__(TPB) void k_write(const float4* __restrict__ x4,
                                               const float* __restrict__ boost,
                                               const unsigned* __restrict__ thresh,
                                               const unsigned* __restrict__ tie_take,
                                               unsigned* __restrict__ tiecnt,
                                               float4* __restrict__ out4) {
    const int tid = threadIdx.x;
    const int batch = blockIdx.y;
    const int slice = blockIdx.x;
    const unsigned T = thresh[batch];
    const unsigned take = tie_take[batch];
    unsigned* cnt = tiecnt + batch;
    const size_t base = (size_t)batch * N4_ + (size_t)slice * V4_PER_BLOCK;
    for (int t = 0; t < TILES; ++t) {
        if (t + 1 < TILES) __builtin_prefetch(&x4[base + (t + 1) * TPB + tid], 0, 1);
        float4 v = x4[base + t * TPB + tid];
        int idx4 = slice * V4_PER_BLOCK + t * TPB + tid;
        float bf = boost[idx4 / (HW_ / 4)];
        float4 r;
        r.x = sel1(v.x, bf, T, take, cnt);
        r.y = sel1(v.y, bf, T, take, cnt);
        r.z = sel1(v.z, bf, T, take, cnt);
        r.w = sel1(v.w, bf, T, take, cnt);
        out4[base + t * TPB + tid] = r;
    }
}

// ---------------------------------------------------------------------------
extern "C" void kernel_launch(void* const* d_in, const int* in_sizes, int n_in,
                              void* d_out, int out_size, void* d_ws, size_t ws_size,
                              hipStream_t stream) {
    (void)in_sizes; (void)n_in; (void)out_size; (void)ws_size;
    const float* x    = (const float*)d_in[0];
    const float* duty = (const float*)d_in[1];
    const int*   kptr = (const int*)d_in[2];

    // workspace carve-up: 64*4096 u32 histograms + per-batch state + boost[C]
    unsigned* hist    = (unsigned*)d_ws;
    unsigned* prefix  = hist + (size_t)B_ * BINS;
    unsigned* needed  = prefix + B_;
    unsigned* thresh  = needed + B_;
    unsigned* tietake = thresh + B_;
    unsigned* tiecnt  = tietake + B_;
    float*    boost   = (float*)(tiecnt + B_);

    const float4* x4   = (const float4*)x;
    float4*       out4 = (float4*)d_out;

    dim3 g(SLICES, B_);
    k_init<<<(B_ * BINS) / TPB, TPB, 0, stream>>>(hist, prefix, needed, tiecnt,
                                                  boost, duty, kptr);
    // level 0: bits [31:20]
    k_hist0<<<g, TPB, 0, stream>>>(x4, boost, hist);
    k_select<<<B_, TPB, 0, stream>>>(hist, prefix, needed, thresh, tietake, tiecnt,
                                     BINS, 20, 0);
    // level 1: bits [19:8] under fixed top-12 prefix
    k_histL<<<g, TPB, 0, stream>>>(x4, boost, prefix, hist, 0xFFF00000u, 8, 0xFFFu);
    k_select<<<B_, TPB, 0, stream>>>(hist, prefix, needed, thresh, tietake, tiecnt,
                                     BINS, 8, 0);
    // level 2: bits [7:0] under fixed top-24 prefix -> exact threshold
    k_histL<<<g, TPB, 0, stream>>>(x4, boost, prefix, hist, 0xFFFFFF00u, 0, 0xFFu);
    k_select<<<B_, TPB, 0, stream>>>(hist, prefix, needed, thresh, tietake, tiecnt,
                                     256, 0, 1);
    // final masked copy of original values
    k_write<<<g, TPB, 0, stream>>>(x4, boost, thresh, tietake, tiecnt, out4);
}